// GNN_node_Virtualnode_21930103014156
// MI455X (gfx1250) — compile-verified
//
#include <hip/hip_runtime.h>

typedef __attribute__((ext_vector_type(2))) float v2f;
typedef __attribute__((ext_vector_type(8))) float v8f;

#define DIM 64

// ---------------- BN folding: scale = g*rsqrt(v+eps), shift = b - m*scale ---------
__global__ void bn_prep_kernel(const float* __restrict__ g, const float* __restrict__ b,
                               const float* __restrict__ m, const float* __restrict__ v,
                               float* __restrict__ scale, float* __restrict__ shift, int n) {
    int i = blockIdx.x * blockDim.x + threadIdx.x;
    if (i < n) {
        float s = g[i] * rsqrtf(v[i] + 1e-5f);
        scale[i] = s;
        shift[i] = b[i] - m[i] * s;
    }
}

// ---------------- vn init: broadcast embedding over B graphs ----------------------
__global__ void vn_init_kernel(const float* __restrict__ emb, float* __restrict__ vn, int total) {
    int i = blockIdx.x * blockDim.x + threadIdx.x;
    if (i < total) vn[i] = emb[i & (DIM - 1)];
}

__global__ void copy_kernel(const float* __restrict__ a, float* __restrict__ b, int n) {
    int i = blockIdx.x * blockDim.x + threadIdx.x;
    if (i < n) b[i] = a[i];
}

// ---------------- per-layer prologue: hcur = hprev + vn[batch], out += hcur, agg=0 -
__global__ void layer_pre_kernel(const float* __restrict__ hprev, const float* __restrict__ vn,
                                 const int* __restrict__ batch_id, float* __restrict__ hcur,
                                 float* __restrict__ outacc, float* __restrict__ agg,
                                 int total, int first) {
    int i = blockIdx.x * blockDim.x + threadIdx.x;
    if (i < total) {
        int n = i >> 6, c = i & 63;
        float v = hprev[i] + vn[batch_id[n] * DIM + c];
        hcur[i] = v;
        outacc[i] = first ? v : (outacc[i] + v);
        agg[i] = 0.0f;
    }
}

// ---------------- edge scatter: agg[dst] += relu(h[src]) -------------------------
__global__ void edge_scatter_kernel(const float* __restrict__ h, const int* __restrict__ src,
                                    const int* __restrict__ dst, float* __restrict__ agg,
                                    long long total) {
    long long i = (long long)blockIdx.x * blockDim.x + threadIdx.x;
    if (i < total) {
        int e = (int)(i >> 4), c = (int)(i & 15);
        int s = src[e], d = dst[e];
        const float4 v = *(const float4*)(h + (size_t)s * DIM + c * 4);
        float* p = agg + (size_t)d * DIM + c * 4;
        atomicAdd(p + 0, fmaxf(v.x, 0.0f));
        atomicAdd(p + 1, fmaxf(v.y, 0.0f));
        atomicAdd(p + 2, fmaxf(v.z, 0.0f));
        atomicAdd(p + 3, fmaxf(v.w, 0.0f));
    }
}

// ---------------- t = (1+eps[l])*hcur + agg (in place into agg) -------------------
__global__ void make_t_kernel(const float* __restrict__ hcur, float* __restrict__ agg,
                              const float* __restrict__ eps, int l, int total) {
    int i = blockIdx.x * blockDim.x + threadIdx.x;
    if (i < total) agg[i] = (1.0f + eps[l]) * hcur[i] + agg[i];
}

// ---------------- vn pooling: run-length reduce sorted batch_id then atomic -------
__global__ void vn_pool_kernel(const float* __restrict__ hcur, const int* __restrict__ batch_id,
                               float* __restrict__ pool, int N) {
    int t = blockIdx.x * blockDim.x + threadIdx.x;
    int c = t & 63;
    int n0 = (t >> 6) * 64;
    if (n0 >= N) return;
    int n1 = (n0 + 64 < N) ? n0 + 64 : N;
    int curb = batch_id[n0];
    float acc = 0.0f;
    for (int n = n0; n < n1; ++n) {
        int b = batch_id[n];
        if (b != curb) {
            atomicAdd(pool + curb * DIM + c, acc);
            acc = 0.0f;
            curb = b;
        }
        acc += hcur[(size_t)n * DIM + c];
    }
    atomicAdd(pool + curb * DIM + c, acc);
}

// ---------------- fp32 WMMA GEMM: out[M,64] = A[M,64] @ W[64,64] + epilogue -------
// One wave32 per 16x16 output tile, K stepped by 4 via V_WMMA_F32_16X16X4_F32.
// A layout (ISA 7.12.2): lane = hi*16+lo holds A[m0+lo][k + 2*hi + {0,1}].
// B layout: vgpr v holds W[k + v + 2*hi][n0+lo] (row = vgpr + 2*hi striping).
// C/D layout: acc[r] = D[m0 + r + 8*hi][n0+lo].
__global__ void wmma_gemm_kernel(const float* __restrict__ A, const float* __restrict__ W,
                                 const float* __restrict__ bias, const float* __restrict__ scale,
                                 const float* __restrict__ shift, float* __restrict__ out,
                                 int M, int relu) {
    int wave = (int)((blockIdx.x * blockDim.x + threadIdx.x) >> 5);
    int lane = threadIdx.x & 31;
    int tiles = (M >> 4) * 4;           // 4 column tiles of 16 over N=64
    if (wave >= tiles) return;          // wave-uniform: EXEC stays all-1s for WMMA
    int tm = wave >> 2, tn = wave & 3;
    int m0 = tm << 4, n0 = tn << 4;
    int lo = lane & 15, hi = lane >> 4;

    const float* arow = A + (size_t)(m0 + lo) * DIM + 2 * hi;
    const float* bcol = W + (size_t)(2 * hi) * DIM + n0 + lo;

    v8f acc = {};
#pragma unroll
    for (int kk = 0; kk < 16; ++kk) {
        v2f a, b;
        a.x = arow[4 * kk + 0];
        a.y = arow[4 * kk + 1];
        b.x = bcol[kk * 4 * DIM];
        b.y = bcol[kk * 4 * DIM + DIM];
        acc = __builtin_amdgcn_wmma_f32_16x16x4_f32(
            /*neg_a=*/false, a, /*neg_b=*/false, b,
            /*c_mod=*/(short)0, acc, /*reuse_a=*/false, /*reuse_b=*/false);
    }

    float bs = bias[n0 + lo];
    bool has_bn = (scale != nullptr);
    float sc = has_bn ? scale[n0 + lo] : 1.0f;
    float sh = has_bn ? shift[n0 + lo] : 0.0f;
#pragma unroll
    for (int r = 0; r < 8; ++r) {
        float v = acc[r] + bs;
        if (has_bn) v = v * sc + sh;
        if (relu) v = fmaxf(v, 0.0f);
        out[(size_t)(m0 + r + 8 * hi) * DIM + n0 + lo] = v;
    }
}

extern "C" void kernel_launch(void* const* d_in, const int* in_sizes, int n_in,
                              void* d_out, int out_size, void* d_ws, size_t ws_size,
                              hipStream_t stream) {
    (void)n_in; (void)out_size; (void)ws_size;

    const float* x      = (const float*)d_in[0];
    const float* lin_W  = (const float*)d_in[1];
    const float* lin_b  = (const float*)d_in[2];
    const float* eps    = (const float*)d_in[3];
    const float* gin_W1 = (const float*)d_in[4];
    const float* gin_b1 = (const float*)d_in[5];
    const float* gbn_g  = (const float*)d_in[6];
    const float* gbn_b  = (const float*)d_in[7];
    const float* gbn_m  = (const float*)d_in[8];
    const float* gbn_v  = (const float*)d_in[9];
    const float* gin_W2 = (const float*)d_in[10];
    const float* gin_b2 = (const float*)d_in[11];
    const float* bn_g   = (const float*)d_in[12];
    const float* bn_b   = (const float*)d_in[13];
    const float* bn_m   = (const float*)d_in[14];
    const float* bn_v   = (const float*)d_in[15];
    const float* vn_emb = (const float*)d_in[16];
    const float* vn_W1  = (const float*)d_in[17];
    const float* vn_b1  = (const float*)d_in[18];
    const float* v1_g   = (const float*)d_in[19];
    const float* v1_b   = (const float*)d_in[20];
    const float* v1_m   = (const float*)d_in[21];
    const float* v1_v   = (const float*)d_in[22];
    const float* vn_W2  = (const float*)d_in[23];
    const float* vn_b2  = (const float*)d_in[24];
    const float* v2_g   = (const float*)d_in[25];
    const float* v2_b   = (const float*)d_in[26];
    const float* v2_m   = (const float*)d_in[27];
    const float* v2_v   = (const float*)d_in[28];
    const int*   src    = (const int*)d_in[29];
    const int*   dst    = (const int*)d_in[30];
    const int*   batch  = (const int*)d_in[31];

    const int N = in_sizes[0] / DIM;      // 50000 (multiple of 16)
    const int L = in_sizes[3];            // 5
    const int E = in_sizes[29];           // 1.6M
    const int B = 128;                    // graphs per batch (multiple of 16)

    float* out = (float*)d_out;

    // ---- workspace layout ----
    float* ws = (float*)d_ws;
    size_t nd = (size_t)N * DIM;
    float* hbuf = ws;                 // h_prev / next-layer h
    float* hcur = hbuf + nd;          // h_list[l] after vn broadcast
    float* agg  = hcur + nd;          // edge aggregation, then t in-place
    float* z1   = agg + nd;           // hidden of GIN MLP
    float* p    = z1 + nd;
    float* vnA   = p; p += (size_t)B * DIM;
    float* vnB   = p; p += (size_t)B * DIM;
    float* pool  = p; p += (size_t)B * DIM;
    float* vtmp  = p; p += (size_t)B * DIM;
    float* g1s = p; p += (size_t)L * DIM;
    float* g1h = p; p += (size_t)L * DIM;
    float* b2s = p; p += (size_t)L * DIM;
    float* b2h = p; p += (size_t)L * DIM;
    float* v1s = p; p += (size_t)(L - 1) * DIM;
    float* v1h = p; p += (size_t)(L - 1) * DIM;
    float* v2s = p; p += (size_t)(L - 1) * DIM;
    float* v2h = p; p += (size_t)(L - 1) * DIM;

    const int TPB = 256;
    auto blocks_for = [](long long n, int tpb) { return (int)((n + tpb - 1) / tpb); };

    // ---- fold all batchnorms ----
    bn_prep_kernel<<<blocks_for(L * DIM, TPB), TPB, 0, stream>>>(gbn_g, gbn_b, gbn_m, gbn_v, g1s, g1h, L * DIM);
    bn_prep_kernel<<<blocks_for(L * DIM, TPB), TPB, 0, stream>>>(bn_g, bn_b, bn_m, bn_v, b2s, b2h, L * DIM);
    bn_prep_kernel<<<blocks_for((L - 1) * DIM, TPB), TPB, 0, stream>>>(v1_g, v1_b, v1_m, v1_v, v1s, v1h, (L - 1) * DIM);
    bn_prep_kernel<<<blocks_for((L - 1) * DIM, TPB), TPB, 0, stream>>>(v2_g, v2_b, v2_m, v2_v, v2s, v2h, (L - 1) * DIM);

    // ---- vn = broadcast(vn_emb) ----
    vn_init_kernel<<<blocks_for(B * DIM, TPB), TPB, 0, stream>>>(vn_emb, vnA, B * DIM);

    auto gemm = [&](const float* Am, const float* Wm, const float* bias, const float* sc,
                    const float* sh, float* o, int M, int relu) {
        int tiles = (M / 16) * 4;                    // one wave per 16x16 tile
        int blocks = (tiles + 7) / 8;                // 8 waves per 256-thread block
        wmma_gemm_kernel<<<blocks, 256, 0, stream>>>(Am, Wm, bias, sc, sh, o, M, relu);
    };

    // ---- input linear: h = x @ lin_W + lin_b ----
    gemm(x, lin_W, lin_b, nullptr, nullptr, hbuf, N, 0);

    long long escatter_total = (long long)E * 16;
    int pool_threads = ((N + 63) / 64) * 64;

    for (int l = 0; l < L; ++l) {
        // hcur = h + vn[batch]; out += hcur; agg = 0
        layer_pre_kernel<<<blocks_for(nd, TPB), TPB, 0, stream>>>(
            hbuf, vnA, batch, hcur, out, agg, (int)nd, l == 0 ? 1 : 0);

        // agg[dst] += relu(hcur[src])
        edge_scatter_kernel<<<blocks_for(escatter_total, TPB), TPB, 0, stream>>>(
            hcur, src, dst, agg, escatter_total);

        // t = (1+eps)*hcur + agg   (in place into agg)
        make_t_kernel<<<blocks_for(nd, TPB), TPB, 0, stream>>>(hcur, agg, eps, l, (int)nd);

        // z1 = relu(bn1(t @ W1 + b1)); h_next = bn2(z1 @ W2 + b2) [+relu if l<L-1]
        gemm(agg, gin_W1 + (size_t)l * DIM * DIM, gin_b1 + l * DIM,
             g1s + l * DIM, g1h + l * DIM, z1, N, 1);
        gemm(z1, gin_W2 + (size_t)l * DIM * DIM, gin_b2 + l * DIM,
             b2s + l * DIM, b2h + l * DIM, hbuf, N, (l < L - 1) ? 1 : 0);

        if (l < L - 1) {
            // pool = vn + segment_sum(hcur, batch)
            copy_kernel<<<blocks_for(B * DIM, TPB), TPB, 0, stream>>>(vnA, pool, B * DIM);
            vn_pool_kernel<<<blocks_for(pool_threads, TPB), TPB, 0, stream>>>(hcur, batch, pool, N);
            // vn MLP (both stages relu(bn(...)))
            gemm(pool, vn_W1 + (size_t)l * DIM * DIM, vn_b1 + l * DIM,
                 v1s + l * DIM, v1h + l * DIM, vtmp, B, 1);
            gemm(vtmp, vn_W2 + (size_t)l * DIM * DIM, vn_b2 + l * DIM,
                 v2s + l * DIM, v2h + l * DIM, vnB, B, 1);
            float* t = vnA; vnA = vnB; vnB = t;      // ping-pong (deterministic per call)
        }
    }
}